// PostprocessWithSampling_6897717477521
// MI455X (gfx1250) — compile-verified
//
#include <hip/hip_runtime.h>
#include <stdint.h>
#include <math.h>

// ---------------------------------------------------------------------------
// Phase 1: partial repetition-penalty + argmax. Each workgroup handles one
// chunk (5 x 1024-element tiles) of one batch row, streaming through LDS with
// a 4-deep pipeline of CDNA5 async global->LDS loads (ASYNCcnt-tracked
// global_load_async_to_lds_b128). Each lane async-loads exactly its own 16B
// of logits and 16B of counts, so there are no cross-lane LDS hazards.
// Partial (bestVal, bestIdx) per chunk goes to workspace; a tiny second phase
// reduces chunks per row with jnp.argmax's lowest-index tie-break.
// ---------------------------------------------------------------------------

#define NTHREADS 256
#define TILE     1024   // NTHREADS lanes * 4 elements
#define DEPTH    4      // LDS pipeline buffers (3 tiles prefetched ahead)

typedef float  f32x4 __attribute__((ext_vector_type(4)));
typedef int    i32x4 __attribute__((ext_vector_type(4)));

__global__ __launch_bounds__(NTHREADS)
void argmax_partial_kernel(const float* __restrict__ logits,
                           const int*   __restrict__ token_count,
                           const float* __restrict__ rep_pen,
                           int V, int chunk, int nparts,
                           float* __restrict__ pvals,
                           int*   __restrict__ pidx)
{
    __shared__ float s_lg[DEPTH][TILE];
    __shared__ int   s_tc[DEPTH][TILE];
    __shared__ float s_wval[NTHREADS / 32];
    __shared__ int   s_widx[NTHREADS / 32];

    const int part = blockIdx.x % nparts;
    const int b    = blockIdx.x / nparts;
    const int tid  = threadIdx.x;
    const float rp = rep_pen[b];

    const int start = part * chunk;
    const int end   = min(start + chunk, V);
    const int n     = max(0, end - start);
    const int tiles = n / TILE;

    const float* lg_row = logits      + (size_t)b * V + start;
    const int*   tc_row = token_count + (size_t)b * V + start;

    float bestV = -__builtin_inff();
    int   bestI = 0x7fffffff;

    // Issue one tile of async global->LDS DMA (2 x b128 per lane).
    auto issue_tile = [&](int t) {
        const unsigned byte_off = (unsigned)((t * TILE + tid * 4) * 4);
        const unsigned lds_lg = (unsigned)(uintptr_t)&s_lg[t & (DEPTH - 1)][tid * 4];
        const unsigned lds_tc = (unsigned)(uintptr_t)&s_tc[t & (DEPTH - 1)][tid * 4];
        asm volatile("global_load_async_to_lds_b128 %0, %1, %2"
                     :: "v"(lds_lg), "v"(byte_off), "s"(lg_row) : "memory");
        asm volatile("global_load_async_to_lds_b128 %0, %1, %2"
                     :: "v"(lds_tc), "v"(byte_off), "s"(tc_row) : "memory");
    };

    const int pre = min(tiles, DEPTH - 1);
    for (int t = 0; t < pre; ++t) issue_tile(t);

    for (int t = 0; t < tiles; ++t) {
        if (t + DEPTH - 1 < tiles) issue_tile(t + DEPTH - 1);
        // Pairs still in flight beyond tile t; async loads complete in order,
        // so waiting for <= 2*ahead guarantees tile t's pair has landed.
        const int ahead = min(tiles - 1 - t, DEPTH - 1);
        switch (ahead) {
            case 3:  asm volatile("s_wait_asynccnt 0x6" ::: "memory"); break;
            case 2:  asm volatile("s_wait_asynccnt 0x4" ::: "memory"); break;
            case 1:  asm volatile("s_wait_asynccnt 0x2" ::: "memory"); break;
            default: asm volatile("s_wait_asynccnt 0x0" ::: "memory"); break;
        }

        const float4 lg4 = *reinterpret_cast<const float4*>(&s_lg[t & (DEPTH - 1)][tid * 4]);
        const int4   tc4 = *reinterpret_cast<const int4*>(&s_tc[t & (DEPTH - 1)][tid * 4]);
        const int e0 = start + t * TILE + tid * 4;

        const float lv[4] = {lg4.x, lg4.y, lg4.z, lg4.w};
        const int   tv[4] = {tc4.x, tc4.y, tc4.z, tc4.w};
        #pragma unroll
        for (int j = 0; j < 4; ++j) {
            const float lg = lv[j];
            float v = lg;
            if (tv[j] > 0) v = (lg < 0.0f) ? lg * rp : lg / rp;  // IEEE div, matches jnp
            const int idx = e0 + j;
            if (v > bestV || (v == bestV && idx < bestI)) { bestV = v; bestI = idx; }
        }
    }

    // Tail (chunk not a multiple of TILE) — guarded direct loads.
    for (int k = tiles * TILE + tid; k < n; k += NTHREADS) {
        const float lg = lg_row[k];
        float v = lg;
        if (tc_row[k] > 0) v = (lg < 0.0f) ? lg * rp : lg / rp;
        const int idx = start + k;
        if (v > bestV || (v == bestV && idx < bestI)) { bestV = v; bestI = idx; }
    }

    // wave32 intra-wave argmax reduction (first-index tie-break)
    #pragma unroll
    for (int off = 16; off > 0; off >>= 1) {
        const float ov = __shfl_down(bestV, off, 32);
        const int   oi = __shfl_down(bestI, off, 32);
        if (ov > bestV || (ov == bestV && oi < bestI)) { bestV = ov; bestI = oi; }
    }
    const int wid  = tid >> 5;
    const int lane = tid & 31;
    if (lane == 0) { s_wval[wid] = bestV; s_widx[wid] = bestI; }
    __syncthreads();
    if (tid == 0) {
        float bv = s_wval[0]; int bi = s_widx[0];
        #pragma unroll
        for (int w = 1; w < NTHREADS / 32; ++w) {
            const float ov = s_wval[w]; const int oi = s_widx[w];
            if (ov > bv || (ov == bv && oi < bi)) { bv = ov; bi = oi; }
        }
        pvals[(size_t)b * nparts + part] = bv;
        pidx [(size_t)b * nparts + part] = bi;
    }
}

// ---------------------------------------------------------------------------
// Kernel 2: streaming int32 -> float32 copy, b128 loads + non-temporal b128
// stores (write-once output should not evict token_count from the 192MB L2),
// with global_prefetch_b8 for the next grid-stride chunk.
// ---------------------------------------------------------------------------
__global__ __launch_bounds__(256)
void i32_to_f32_kernel(const int* __restrict__ src, float* __restrict__ dst,
                       long long n)
{
    const long long n4     = n >> 2;
    const long long start  = (long long)blockIdx.x * blockDim.x + threadIdx.x;
    const long long stride = (long long)gridDim.x * blockDim.x;
    const i32x4* s4 = reinterpret_cast<const i32x4*>(src);
    f32x4*       d4 = reinterpret_cast<f32x4*>(dst);
    for (long long i = start; i < n4; i += stride) {
        __builtin_prefetch(&s4[i + stride], 0, 0);   // global_prefetch_b8
        const i32x4 v = s4[i];
        const f32x4 f = __builtin_convertvector(v, f32x4);
        __builtin_nontemporal_store(f, &d4[i]);      // TH=NT b128 store
    }
    for (long long j = (n4 << 2) + start; j < n; j += stride)
        dst[j] = (float)src[j];
}

// ---------------------------------------------------------------------------
// Kernel 3: final per-row argmax reduction over chunk partials + all scalar
// fixups (runs after the bulk copies).
// d_out layout (floats): tokens[B] | lti[B] | attn[B*L] | gent[B*L] |
//                        gen_idx[B] | token_count[B*V]
// ---------------------------------------------------------------------------
__global__ void reduce_fixup_kernel(const float* __restrict__ pvals,
                                    const int*   __restrict__ pidx,
                                    const int*   __restrict__ last_token_index,
                                    const int*   __restrict__ generated_index,
                                    int B, int V, int L, int nparts,
                                    float* __restrict__ out)
{
    const int b = blockIdx.x * blockDim.x + threadIdx.x;
    if (b >= B) return;

    float bv = pvals[(size_t)b * nparts];
    int   bi = pidx [(size_t)b * nparts];
    for (int p = 1; p < nparts; ++p) {
        const float ov = pvals[(size_t)b * nparts + p];
        const int   oi = pidx [(size_t)b * nparts + p];
        if (ov > bv || (ov == bv && oi < bi)) { bv = ov; bi = oi; }
    }
    const int tok = bi;

    float* out_tok  = out;
    float* out_lti  = out + B;
    float* out_attn = out + 2 * (size_t)B;
    float* out_gent = out_attn + (size_t)B * L;
    float* out_gidx = out_gent + (size_t)B * L;
    float* out_tc   = out_gidx + B;

    const int lti = min(last_token_index[b] + 1, L - 1);
    const int gi  = generated_index[b];

    out_tok[b] = (float)tok;
    out_lti[b] = (float)lti;
    out_attn[(size_t)b * L + lti] = 1.0f;
    out_gent[(size_t)b * L + gi]  = (float)tok;
    out_gidx[b] = (float)min(gi + 1, L - 1);
    out_tc[(size_t)b * V + tok] += 1.0f;   // one thread per row: safe RMW
}

// ---------------------------------------------------------------------------
extern "C" void kernel_launch(void* const* d_in, const int* in_sizes, int n_in,
                              void* d_out, int out_size, void* d_ws, size_t ws_size,
                              hipStream_t stream)
{
    const float* logits  = (const float*)d_in[0];  // (B,1,V) f32
    const int*   lti_in  = (const int*)  d_in[1];  // (B,1)   i32
    const int*   attn_in = (const int*)  d_in[2];  // (B,L)   i32
    const int*   gent_in = (const int*)  d_in[3];  // (B,L)   i32
    const int*   gidx_in = (const int*)  d_in[4];  // (B,1)   i32
    const float* rep     = (const float*)d_in[5];  // (B,1)   f32
    const int*   tc_in   = (const int*)  d_in[6];  // (B,V)   i32

    const int B = in_sizes[1];            // 128
    const int V = in_sizes[0] / B;        // 128000
    const int L = in_sizes[2] / B;        // 8192

    // Split each row into ~25 chunks rounded to whole 1024-elem tiles:
    // 128000 -> chunk=5120 (5 tiles exactly), nparts=25, grid = 3200 WGs.
    int chunk = (V + 24) / 25;
    chunk = ((chunk + TILE - 1) / TILE) * TILE;
    if (chunk < TILE) chunk = TILE;
    const int nparts = (V + chunk - 1) / chunk;

    float* out = (float*)d_out;
    float* out_attn = out + 2 * (size_t)B;
    float* out_gent = out_attn + (size_t)B * L;
    float* out_tc   = out_gent + (size_t)B * L + B;

    float* pvals = (float*)d_ws;
    int*   pidx  = (int*)(pvals + (size_t)B * nparts);

    // 1) partial penalty+argmax, 25 WGs per batch row
    argmax_partial_kernel<<<B * nparts, NTHREADS, 0, stream>>>(
        logits, tc_in, rep, V, chunk, nparts, pvals, pidx);

    // 2) bulk int->float copies of pass-through outputs (NT stores)
    i32_to_f32_kernel<<<2048, 256, 0, stream>>>(attn_in, out_attn, (long long)B * L);
    i32_to_f32_kernel<<<2048, 256, 0, stream>>>(gent_in, out_gent, (long long)B * L);
    i32_to_f32_kernel<<<4096, 256, 0, stream>>>(tc_in,   out_tc,   (long long)B * V);

    // 3) final reduce + scalar scatters
    reduce_fixup_kernel<<<(B + 127) / 128, 128, 0, stream>>>(
        pvals, pidx, lti_in, gidx_in, B, V, L, nparts, out);
}